// GCN_with_dropout_77721728189011
// MI455X (gfx1250) — compile-verified
//
#include <hip/hip_runtime.h>
#include <cstdint>

typedef __attribute__((ext_vector_type(2))) float v2f;
typedef __attribute__((ext_vector_type(8))) float v8f;

// ---------------------------------------------------------------------------
// 1) degree: atomic count of dst occurrences, then dis = rsqrt(deg + 1)
// ---------------------------------------------------------------------------
__global__ void deg_count_kernel(const int* __restrict__ dst, int E,
                                 float* __restrict__ deg) {
    int e = blockIdx.x * blockDim.x + threadIdx.x;
    if (e < E) atomicAdd(&deg[dst[e]], 1.0f);
}

__global__ void deg_finalize_kernel(float* __restrict__ dis, int n) {
    int i = blockIdx.x * blockDim.x + threadIdx.x;
    if (i < n) dis[i] = rsqrtf(dis[i] + 1.0f);
}

// ---------------------------------------------------------------------------
// 2) dense GEMM  C[M,NC] = A[M,K] * B[K,NC]  via V_WMMA_F32_16X16X4_F32.
//    One wave32 per 16x16 output tile.
//    fp32 A 16x4 fragment (ISA 7.12.2): lanes 0-15 hold row M=lane, {K+0,K+1};
//    lanes 16-31 hold row M=lane-16, {K+2,K+3}. B fragment mirrors along N.
//    fp32 C/D 16x16: 8 VGPRs; lane group selects M base 0 or 8, N = lane&15.
// ---------------------------------------------------------------------------
template <int K, int NC>
__global__ void gemm_wmma_f32(const float* __restrict__ A,
                              const float* __restrict__ B,
                              float* __restrict__ C, int nTiles) {
    int wave = (int)((blockIdx.x * blockDim.x + threadIdx.x) >> 5);
    int lane = threadIdx.x & 31;
    if (wave >= nTiles) return;                 // wave-uniform: EXEC stays all-1s
    constexpr int TN = NC / 16;
    int tm = wave / TN;
    int tn = wave % TN;
    int rowBase = tm * 16;
    int colBase = tn * 16;
    int hi = lane >> 4;                          // 0: K+{0,1}, 1: K+{2,3}
    int l  = lane & 15;

    const float* arow = A + (size_t)(rowBase + l) * K;
    const float* bcol = B + colBase + l;

    v8f acc = {};
#pragma unroll
    for (int k = 0; k < K; k += 4) {
        v2f a, b;
        a.x = arow[k + 2 * hi + 0];
        a.y = arow[k + 2 * hi + 1];
        b.x = bcol[(size_t)(k + 2 * hi + 0) * NC];
        b.y = bcol[(size_t)(k + 2 * hi + 1) * NC];
        acc = __builtin_amdgcn_wmma_f32_16x16x4_f32(
            /*neg_a=*/false, a, /*neg_b=*/false, b,
            /*c_mod=*/(short)0, acc, /*reuse_a=*/false, /*reuse_b=*/false);
    }

    float* cptr = C + (size_t)colBase + l;
#pragma unroll
    for (int i = 0; i < 8; ++i)
        cptr[(size_t)(rowBase + 8 * hi + i) * NC] = acc[i];
}

// ---------------------------------------------------------------------------
// 3) edge aggregation: agg[dst] += h[src] * dis[src]*dis[dst]  (F feats/edge)
//    One thread per (edge, feature); atomics land in L2 (h + agg fit in 192MB).
// ---------------------------------------------------------------------------
template <int LOGF>
__global__ void edge_agg_kernel(const int* __restrict__ src,
                                const int* __restrict__ dst,
                                const float* __restrict__ dis,
                                const float* __restrict__ h,
                                float* __restrict__ agg, long long total) {
    long long t = (long long)blockIdx.x * blockDim.x + threadIdx.x;
    if (t >= total) return;
    int e = (int)(t >> LOGF);
    int j = (int)(t & ((1 << LOGF) - 1));
    int s = src[e];
    int d = dst[e];
    float norm = dis[s] * dis[d];
    float v = h[((size_t)s << LOGF) + j] * norm;
    atomicAdd(&agg[((size_t)d << LOGF) + j], v);
}

// ---------------------------------------------------------------------------
// 4) layer-1 epilogue (in place on agg1): relu(agg + h*dis^2 + bias)
// ---------------------------------------------------------------------------
__global__ void selfloop_bias_relu_kernel(const float* __restrict__ h,
                                          const float* __restrict__ dis,
                                          const float* __restrict__ bias,
                                          float* __restrict__ agg, int n) {
    int t = blockIdx.x * blockDim.x + threadIdx.x;
    if (t >= n) return;
    int i = t >> 6;               // row   (F = 64)
    int j = t & 63;               // feat
    float dii = dis[i];
    float v = agg[t] + h[t] * dii * dii + bias[j];
    agg[t] = fmaxf(v, 0.0f);
}

// ---------------------------------------------------------------------------
// 5) layer-2 epilogue + log_softmax over 16 features, one thread per node
// ---------------------------------------------------------------------------
__global__ void logsoftmax_kernel(const float* __restrict__ h,
                                  const float* __restrict__ dis,
                                  const float* __restrict__ bias,
                                  const float* __restrict__ agg,
                                  float* __restrict__ out, int n) {
    int i = blockIdx.x * blockDim.x + threadIdx.x;
    if (i >= n) return;
    float dii = dis[i];
    dii *= dii;
    float v[16];
    float mx = -3.402823466e38f;
#pragma unroll
    for (int j = 0; j < 16; ++j) {
        size_t idx = (size_t)i * 16 + j;
        v[j] = agg[idx] + h[idx] * dii + bias[j];
        mx = fmaxf(mx, v[j]);
    }
    float s = 0.0f;
#pragma unroll
    for (int j = 0; j < 16; ++j) s += __expf(v[j] - mx);
    float lse = mx + __logf(s);
#pragma unroll
    for (int j = 0; j < 16; ++j) out[(size_t)i * 16 + j] = v[j] - lse;
}

// ---------------------------------------------------------------------------
extern "C" void kernel_launch(void* const* d_in, const int* in_sizes, int n_in,
                              void* d_out, int out_size, void* d_ws, size_t ws_size,
                              hipStream_t stream) {
    const float* x   = (const float*)d_in[0];   // [N,128]
    const int*   ei  = (const int*)  d_in[1];   // [2,E]
    const float* W1  = (const float*)d_in[2];   // [128,64]
    const float* b1  = (const float*)d_in[3];   // [64]
    const float* W2  = (const float*)d_in[4];   // [64,16]
    const float* b2  = (const float*)d_in[5];   // [16]
    float* out = (float*)d_out;

    const int N = in_sizes[0] / 128;            // 100000 (divisible by 16)
    const int E = in_sizes[1] / 2;              // 1600000
    const int* src = ei;
    const int* dst = ei + E;

    // workspace layout (floats): dis[N] | h1[N*64] | agg1[N*64] | h2[N*16] | agg2[N*16]
    float* dis  = (float*)d_ws;
    float* h1   = dis  + N;
    float* agg1 = h1   + (size_t)N * 64;
    float* h2   = agg1 + (size_t)N * 64;
    float* agg2 = h2   + (size_t)N * 16;

    hipMemsetAsync(dis,  0, (size_t)N * sizeof(float), stream);
    hipMemsetAsync(agg1, 0, (size_t)N * 64 * sizeof(float), stream);
    hipMemsetAsync(agg2, 0, (size_t)N * 16 * sizeof(float), stream);

    const int T = 256;

    // degrees -> dis = rsqrt(deg + 1)
    deg_count_kernel<<<(E + T - 1) / T, T, 0, stream>>>(dst, E, dis);
    deg_finalize_kernel<<<(N + T - 1) / T, T, 0, stream>>>(dis, N);

    // layer 1: h1 = x @ W1
    {
        int nTiles = (N / 16) * (64 / 16);
        int waves_per_block = T / 32;
        gemm_wmma_f32<128, 64><<<(nTiles + waves_per_block - 1) / waves_per_block,
                                 T, 0, stream>>>(x, W1, h1, nTiles);
    }
    // scatter: agg1[dst] += h1[src] * norm
    {
        long long total = (long long)E * 64;
        edge_agg_kernel<6><<<(int)((total + T - 1) / T), T, 0, stream>>>(
            src, dst, dis, h1, agg1, total);
    }
    // agg1 = relu(agg1 + h1*dis^2 + b1)   (in place -> becomes layer-2 input)
    {
        int total = N * 64;
        selfloop_bias_relu_kernel<<<(total + T - 1) / T, T, 0, stream>>>(
            h1, dis, b1, agg1, total);
    }
    // layer 2: h2 = agg1 @ W2
    {
        int nTiles = (N / 16) * (16 / 16);
        int waves_per_block = T / 32;
        gemm_wmma_f32<64, 16><<<(nTiles + waves_per_block - 1) / waves_per_block,
                                T, 0, stream>>>(agg1, W2, h2, nTiles);
    }
    // scatter: agg2[dst] += h2[src] * norm
    {
        long long total = (long long)E * 16;
        edge_agg_kernel<4><<<(int)((total + T - 1) / T), T, 0, stream>>>(
            src, dst, dis, h2, agg2, total);
    }
    // out = log_softmax(agg2 + h2*dis^2 + b2)
    logsoftmax_kernel<<<(N + T - 1) / T, T, 0, stream>>>(h2, dis, b2, agg2, out, N);
}